// SectorAwareFusionModel_1005022347506
// MI455X (gfx1250) — compile-verified
//
#include <hip/hip_runtime.h>

// Problem sizes (fixed by the reference)
#define B_TOTAL   131072
#define D_FUSED   256
#define D_SHARED  128
#define H_HEAD    64
#define N_CLASSES 3
#define N_SECTORS 13

// Stage-1 GEMM: 4 waves/block, each wave owns 32 rows (two 16-row C tiles)
#define G1_WAVES       4
#define G1_ROWS_BLOCK  (G1_WAVES * 32)
#define G1_THREADS     (G1_WAVES * 32)

// Stage-2 head: 4 waves/block, one sector-pure 16-row tile per wave
#define H_WAVES        4
#define H_THREADS      (H_WAVES * 32)
// worst-case tiles = B/16 + (N_SECTORS-1) = 8204; round to multiple of H_WAVES
#define MAX_TILES      8208
#define PERM_INTS      (MAX_TILES * 16)

typedef __attribute__((ext_vector_type(2))) float v2f;
typedef __attribute__((ext_vector_type(8))) float v8f;

// D = A(16x4) * B(4x16) + C(16x16), fp32 matrix pipe on gfx1250.
__device__ __forceinline__ v8f wmma_f32_16x16x4(v2f a, v2f b, v8f c) {
    return __builtin_amdgcn_wmma_f32_16x16x4_f32(
        /*neg_a=*/false, a, /*neg_b=*/false, b,
        /*c_mod=*/(short)0, c, /*reuse_a=*/false, /*reuse_b=*/false);
}

// ---------------------------------------------------------------------------
// Counting sort of rows by sector (13 buckets), block-aggregated atomics.
// ---------------------------------------------------------------------------
__global__ __launch_bounds__(256)
void count_kernel(const int* __restrict__ sector_id, int* __restrict__ counts)
{
    __shared__ int lc[N_SECTORS];
    if (threadIdx.x < N_SECTORS) lc[threadIdx.x] = 0;
    __syncthreads();
    const int row = blockIdx.x * 256 + threadIdx.x;
    atomicAdd(&lc[sector_id[row]], 1);
    __syncthreads();
    if (threadIdx.x < N_SECTORS) atomicAdd(&counts[threadIdx.x], lc[threadIdx.x]);
}

__global__ __launch_bounds__(32)
void base_kernel(const int* __restrict__ counts, int* __restrict__ padded_base)
{
    if (threadIdx.x == 0) {
        int run = 0;
        for (int s = 0; s < N_SECTORS; ++s) {
            padded_base[s] = run;
            run += ((counts[s] + 15) >> 4) << 4;   // pad each bucket to 16 rows
        }
    }
}

__global__ __launch_bounds__(256)
void scatter_kernel(const int* __restrict__ sector_id,
                    const int* __restrict__ padded_base,
                    int* __restrict__ cursor, int* __restrict__ perm)
{
    __shared__ int lc[N_SECTORS];
    __shared__ int lbase[N_SECTORS];
    if (threadIdx.x < N_SECTORS) lc[threadIdx.x] = 0;
    __syncthreads();
    const int row = blockIdx.x * 256 + threadIdx.x;
    const int sec = sector_id[row];
    const int lpos = atomicAdd(&lc[sec], 1);
    __syncthreads();
    if (threadIdx.x < N_SECTORS)
        lbase[threadIdx.x] = atomicAdd(&cursor[threadIdx.x], lc[threadIdx.x]);
    __syncthreads();
    perm[padded_base[sec] + lbase[sec] + lpos] = row;
}

// ---------------------------------------------------------------------------
// Stage 1: shared = relu(fused @ Wf + bf).  Pure WMMA, no LDS.
// Each wave: 32 rows (two 16-row tiles) so every B-frag feeds 2 WMMAs.
// ---------------------------------------------------------------------------
__global__ __launch_bounds__(G1_THREADS)
void fusion_gemm_kernel(const float* __restrict__ fused,
                        const float* __restrict__ Wf,
                        const float* __restrict__ bf,
                        float* __restrict__ out_shared)
{
    const int tid  = threadIdx.x;
    const int w    = tid >> 5;
    const int lane = tid & 31;
    const int m    = lane & 15;
    const int half = lane >> 4;
    const int row0 = blockIdx.x * G1_ROWS_BLOCK + w * 32;

    v8f acc0[8] = {};
    v8f acc1[8] = {};

    const float* ap0 = fused + (size_t)(row0 + m) * D_FUSED;
    const float* ap1 = ap0 + (size_t)16 * D_FUSED;

    for (int k0 = 0; k0 < D_FUSED; k0 += 4) {
        const int kb = k0 + 2 * half;
        const v2f a0 = *(const v2f*)(ap0 + kb);
        const v2f a1 = *(const v2f*)(ap1 + kb);
        const float* wrow = Wf + (size_t)kb * D_SHARED + m;
        #pragma unroll
        for (int n = 0; n < 8; ++n) {
            v2f b;
            b.x = wrow[n * 16];
            b.y = wrow[D_SHARED + n * 16];
            acc0[n] = wmma_f32_16x16x4(a0, b, acc0[n]);
            acc1[n] = wmma_f32_16x16x4(a1, b, acc1[n]);
        }
    }

    #pragma unroll
    for (int n = 0; n < 8; ++n) {
        const int col = n * 16 + m;
        const float bv = bf[col];
        #pragma unroll
        for (int r = 0; r < 8; ++r) {
            const int rl = r + 8 * half;
            out_shared[(size_t)(row0 + rl) * D_SHARED + col] =
                fmaxf(acc0[n][r] + bv, 0.0f);
            out_shared[(size_t)(row0 + 16 + rl) * D_SHARED + col] =
                fmaxf(acc1[n][r] + bv, 0.0f);
        }
    }
}

// ---------------------------------------------------------------------------
// Stage 2: sector-pure head tiles.  Wave t gathers rows perm[t*16..t*16+15]
// (all the same sector s), computes h = relu(shared @ W1[s] + b1[s]) with
// WMMA (K=128), then y = h @ W2[s] + b2[s] and softmax per row.
// ---------------------------------------------------------------------------
__global__ __launch_bounds__(H_THREADS)
void sector_head_kernel(const float* __restrict__ shared_feat,
                        const float* __restrict__ W1,
                        const float* __restrict__ b1,
                        const float* __restrict__ W2,
                        const float* __restrict__ b2,
                        const int*   __restrict__ sector_id,
                        const int*   __restrict__ perm,
                        float* __restrict__ out_logits,
                        float* __restrict__ out_probs)
{
    __shared__ float lds_h[H_WAVES][16][H_HEAD];   // 16 KB

    const int tid  = threadIdx.x;
    const int w    = tid >> 5;
    const int lane = tid & 31;
    const int m    = lane & 15;
    const int half = lane >> 4;
    const int t    = blockIdx.x * H_WAVES + w;

    const int* ptile = perm + t * 16;
    const int row_first = ptile[0];          // lane-uniform
    if (row_first < 0) return;               // padding / beyond-last tile: whole wave exits

    const int  myrow = ptile[m];             // lanes l and l+16 share a row
    const bool valid = (myrow >= 0);
    const int  arow  = valid ? myrow : row_first;  // duplicate a real row for pad lanes
    const int  s     = sector_id[row_first];

    // h accumulators seeded with b1[s]
    v8f h[4];
    #pragma unroll
    for (int n = 0; n < 4; ++n) {
        const float b1v = b1[s * H_HEAD + n * 16 + m];
        #pragma unroll
        for (int r = 0; r < 8; ++r) h[n][r] = b1v;
    }

    const float* ap  = shared_feat + (size_t)arow * D_SHARED;
    const float* w1s = W1 + (size_t)s * D_SHARED * H_HEAD;
    for (int k0 = 0; k0 < D_SHARED; k0 += 4) {
        const int kb = k0 + 2 * half;
        const v2f a = *(const v2f*)(ap + kb);
        #pragma unroll
        for (int n = 0; n < 4; ++n) {
            v2f b;
            b.x = w1s[(size_t)kb * H_HEAD + n * 16 + m];
            b.y = w1s[(size_t)(kb + 1) * H_HEAD + n * 16 + m];
            h[n] = wmma_f32_16x16x4(a, b, h[n]);
        }
    }

    // relu + stage h tile in per-wave LDS (C layout -> row major)
    #pragma unroll
    for (int n = 0; n < 4; ++n) {
        #pragma unroll
        for (int r = 0; r < 8; ++r)
            lds_h[w][r + 8 * half][n * 16 + m] = fmaxf(h[n][r], 0.0f);
    }
    // DS ops are in-order within a wave; fence the compiler + wait ds traffic
    asm volatile("s_wait_dscnt 0x0" ::: "memory");

    // y = h @ W2[s] : two lanes per row, K split by half
    float y0 = 0.f, y1 = 0.f, y2 = 0.f;
    {
        const float* w2s  = W2 + (size_t)s * H_HEAD * N_CLASSES;
        const int    koff = half * 32;
        #pragma unroll 8
        for (int k = 0; k < 32; ++k) {
            const float hv = lds_h[w][m][koff + k];
            y0 += hv * w2s[(koff + k) * 3 + 0];
            y1 += hv * w2s[(koff + k) * 3 + 1];
            y2 += hv * w2s[(koff + k) * 3 + 2];
        }
    }
    y0 += __shfl_xor(y0, 16, 32);
    y1 += __shfl_xor(y1, 16, 32);
    y2 += __shfl_xor(y2, 16, 32);

    if (valid && half == 0) {
        y0 += b2[s * 3 + 0];
        y1 += b2[s * 3 + 1];
        y2 += b2[s * 3 + 2];
        const float mx = fmaxf(y0, fmaxf(y1, y2));
        const float e0 = __expf(y0 - mx);
        const float e1 = __expf(y1 - mx);
        const float e2 = __expf(y2 - mx);
        const float inv = 1.0f / (e0 + e1 + e2);
        out_logits[(size_t)myrow * 3 + 0] = y0;
        out_logits[(size_t)myrow * 3 + 1] = y1;
        out_logits[(size_t)myrow * 3 + 2] = y2;
        out_probs[(size_t)myrow * 3 + 0] = e0 * inv;
        out_probs[(size_t)myrow * 3 + 1] = e1 * inv;
        out_probs[(size_t)myrow * 3 + 2] = e2 * inv;
    }
}

// ---------------------------------------------------------------------------
extern "C" void kernel_launch(void* const* d_in, const int* in_sizes, int n_in,
                              void* d_out, int out_size, void* d_ws, size_t ws_size,
                              hipStream_t stream) {
    const float* fused     = (const float*)d_in[0];
    const float* Wf        = (const float*)d_in[1];
    const float* bf        = (const float*)d_in[2];
    const float* W1        = (const float*)d_in[3];
    const float* b1        = (const float*)d_in[4];
    const float* W2        = (const float*)d_in[5];
    const float* b2        = (const float*)d_in[6];
    const int*   sector_id = (const int*)d_in[7];

    float* out        = (float*)d_out;
    float* out_logits = out;                                   // [B,3]
    float* out_probs  = out + (size_t)B_TOTAL * N_CLASSES;     // [B,3]
    float* out_shared = out + (size_t)B_TOTAL * 2 * N_CLASSES; // [B,128]

    // workspace layout (ints): [0..13) counts | [16..29) padded_base |
    // [32..45) cursor | [64 ..) perm (PERM_INTS)
    int* ws_i        = (int*)d_ws;
    int* counts      = ws_i;
    int* padded_base = ws_i + 16;
    int* cursor      = ws_i + 32;
    int* perm        = ws_i + 64;

    hipMemsetAsync(ws_i, 0, 64 * sizeof(int), stream);
    hipMemsetAsync(perm, 0xFF, (size_t)PERM_INTS * sizeof(int), stream); // -1 pad marker

    count_kernel<<<dim3(B_TOTAL / 256), dim3(256), 0, stream>>>(sector_id, counts);
    base_kernel<<<dim3(1), dim3(32), 0, stream>>>(counts, padded_base);
    scatter_kernel<<<dim3(B_TOTAL / 256), dim3(256), 0, stream>>>(
        sector_id, padded_base, cursor, perm);

    fusion_gemm_kernel<<<dim3(B_TOTAL / G1_ROWS_BLOCK), dim3(G1_THREADS), 0, stream>>>(
        fused, Wf, bf, out_shared);

    sector_head_kernel<<<dim3(MAX_TILES / H_WAVES), dim3(H_THREADS), 0, stream>>>(
        out_shared, W1, b1, W2, b2, sector_id, perm, out_logits, out_probs);
}